// GCNModel_44762149159246
// MI455X (gfx1250) — compile-verified
//
#include <hip/hip_runtime.h>
#include <stdint.h>

// GCN forward for MI455X (gfx1250), fp32 WMMA + async global->LDS staging.
// B=4096, N=128, F_IN=67, H=64, L=4. One workgroup (8 wave32) per batch element.

typedef __attribute__((ext_vector_type(2))) float v2f;
typedef __attribute__((ext_vector_type(8))) float v8f;

#define N_SZ   128
#define F_IN_  67
#define H_SZ   64
#define L_SZ   4
// Row pitches chosen so pitch % 64 == 4 (floats): the two half-wave K-offsets
// (+0 / +2) of a b64 fragment load hit disjoint bank pairs -> conflict-free.
#define ADJ_LD 132   // adj_s[row 0..127][k 0..127]
#define H_LD   68    // h_s  [row 0..127][k 0..63]
#define HWT_LD 132   // hw_t [n 0..63][k 0..127]   (transposed)
#define WT_LD  68    // w_t  [n 0..63][k 0..67]    (transposed)

#define SMEM_FLOATS (N_SZ*ADJ_LD + N_SZ*H_LD + H_SZ*HWT_LD + H_SZ*WT_LD + 64)

__device__ __forceinline__ v8f wmma4(v2f a, v2f b, v8f c) {
  // V_WMMA_F32_16X16X4_F32 : D = A(16x4) * B(4x16) + C(16x16), all f32
  return __builtin_amdgcn_wmma_f32_16x16x4_f32(
      false, a, false, b, (short)0, c, false, false);
}

__device__ __forceinline__ float sigmoidf_(float x) {
  return 1.0f / (1.0f + __expf(-x));
}

__launch_bounds__(256, 1)
__global__ void gcn_fused_kernel(const float* __restrict__ node,
                                 const float* __restrict__ adj,
                                 const float* __restrict__ W_emb,
                                 const float* __restrict__ W_gcn,
                                 const float* __restrict__ b_gcn,
                                 const float* __restrict__ W_fc,
                                 const float* __restrict__ b_fc,
                                 float* __restrict__ out) {
  extern __shared__ float smem[];
  float* adj_s = smem;                         // 128 x 132
  float* h_s   = adj_s + N_SZ * ADJ_LD;        // 128 x 68
  float* hw_t  = h_s   + N_SZ * H_LD;          // 64 x 132  (hw transposed)
  float* w_t   = hw_t  + H_SZ * HWT_LD;        // 64 x 68   (weights transposed)
  float* red_s = w_t   + H_SZ * WT_LD;         // 64

  const int b    = blockIdx.x;
  const int tid  = threadIdx.x;
  const int lane = tid & 31;
  const int wv   = tid >> 5;      // wave 0..7 -> row stripe 16*wv .. 16*wv+15
  const int half = lane >> 4;     // 0/1: K (A,B frags) or M (C frag) split
  const int l15  = lane & 15;

  // ---------- async-stage adj[b] (128x128, 64 KB) global -> LDS ----------
  // GLOBAL_LOAD_ASYNC_TO_LDS_B128, GVS mode: saddr = adj[b] matrix base (uniform),
  // vaddr = per-lane byte offset, vdst = per-lane LDS byte address.
  const float* adj_g = adj + (size_t)b * N_SZ * N_SZ;
  {
    const unsigned lds_base = (unsigned)(uintptr_t)adj_s;
    #pragma unroll
    for (int i = 0; i < 16; ++i) {
      const int idx = tid + i * 256;          // 4096 x b128 total
      const int row = idx >> 5;               // 32 x b128 per row
      const int c4  = idx & 31;
      const unsigned ldsoff = lds_base + (unsigned)(row * ADJ_LD + c4 * 4) * 4u;
      const unsigned goff   = (unsigned)(row * N_SZ + c4 * 4) * 4u;
      asm volatile("global_load_async_to_lds_b128 %0, %1, %2"
                   :: "v"(ldsoff), "v"(goff), "s"(adj_g) : "memory");
    }
  }

  // ---------- stage W_emb transposed into w_t (k=67 zero-padded) ----------
  for (int idx = tid; idx < F_IN_ * H_SZ; idx += 256) {
    const int k = idx >> 6;                   // 0..66
    const int n = idx & 63;
    w_t[n * WT_LD + k] = W_emb[idx];
  }
  if (tid < H_SZ) w_t[tid * WT_LD + F_IN_] = 0.0f;   // k = 67 pad
  __syncthreads();   // w_t ready (adj still in flight)

  // ---------- embedding: h = node[b](128x67) @ W_emb(67x64) ----------
  {
    const float* node_g = node + (size_t)b * N_SZ * F_IN_;
    const int m = wv * 16 + l15;              // A-frag row for this lane
    v8f acc[4];
    #pragma unroll
    for (int t = 0; t < 4; ++t) { v8f z = {}; acc[t] = z; }
    for (int kk = 0; kk < 68; kk += 4) {      // K padded 67 -> 68
      const int k0 = kk + 2 * half;           // k0 <= 66: a.x always valid
      v2f a;
      a.x = node_g[m * F_IN_ + k0];
      a.y = (k0 + 1 < F_IN_) ? node_g[m * F_IN_ + k0 + 1] : 0.0f;
      #pragma unroll
      for (int t = 0; t < 4; ++t) {
        const int n = t * 16 + l15;
        const v2f bb = *(const v2f*)(w_t + n * WT_LD + k0);  // ds_load_b64
        acc[t] = wmma4(a, bb, acc[t]);
      }
    }
    #pragma unroll
    for (int t = 0; t < 4; ++t) {
      #pragma unroll
      for (int r = 0; r < 8; ++r) {
        const int row = wv * 16 + r + 8 * half;
        const int col = t * 16 + l15;
        h_s[row * H_LD + col] = acc[t][r];
      }
    }
  }
  // adj async copies must have landed before anyone reads adj_s
  asm volatile("s_wait_asynccnt 0x0" ::: "memory");
  __syncthreads();   // h_s + adj_s ready; emb reads of w_t finished

  // ---------- 4 GCN layers ----------
  for (int li = 0; li < L_SZ; ++li) {
    const float* Wl = W_gcn + li * H_SZ * H_SZ;
    const float* bl = b_gcn + li * H_SZ;

    // stage Wl transposed: w_t[n][k] = Wl[k][n]
    #pragma unroll
    for (int i = 0; i < 16; ++i) {
      const int idx = tid + i * 256;          // 4096 elements
      const int k = idx >> 6;
      const int n = idx & 63;
      w_t[n * WT_LD + k] = Wl[idx];
    }
    __syncthreads();  // w_t ready

    // hw = h @ Wl + bl (128x64 @ 64x64), stored transposed into hw_t
    {
      v8f acc[4];
      #pragma unroll
      for (int t = 0; t < 4; ++t) { v8f z = {}; acc[t] = z; }
      for (int kk = 0; kk < H_SZ; kk += 4) {
        const int k0 = kk + 2 * half;
        const v2f a = *(const v2f*)(h_s + (wv * 16 + l15) * H_LD + k0);
        #pragma unroll
        for (int t = 0; t < 4; ++t) {
          const int n = t * 16 + l15;
          const v2f bb = *(const v2f*)(w_t + n * WT_LD + k0);
          acc[t] = wmma4(a, bb, acc[t]);
        }
      }
      #pragma unroll
      for (int t = 0; t < 4; ++t) {
        const float bias = bl[t * 16 + l15];
        #pragma unroll
        for (int r = 0; r < 8; ++r) {
          const int n   = t * 16 + l15;               // hw column
          const int row = wv * 16 + r + 8 * half;     // hw row -> hw_t col
          hw_t[n * HWT_LD + row] = acc[t][r] + bias;
        }
      }
    }
    __syncthreads();  // hw_t ready

    // msg = adj @ hw (128x128 @ 128x64); h = relu(sigmoid(msg) + h)
    {
      v8f acc[4];
      #pragma unroll
      for (int t = 0; t < 4; ++t) { v8f z = {}; acc[t] = z; }
      for (int kk = 0; kk < N_SZ; kk += 4) {
        const int k0 = kk + 2 * half;
        const v2f a = *(const v2f*)(adj_s + (wv * 16 + l15) * ADJ_LD + k0);
        #pragma unroll
        for (int t = 0; t < 4; ++t) {
          const int n = t * 16 + l15;
          const v2f bb = *(const v2f*)(hw_t + n * HWT_LD + k0);
          acc[t] = wmma4(a, bb, acc[t]);
        }
      }
      // fused sigmoid + residual + relu; rows are wave-private
      #pragma unroll
      for (int t = 0; t < 4; ++t) {
        #pragma unroll
        for (int r = 0; r < 8; ++r) {
          const int row = wv * 16 + r + 8 * half;
          const int col = t * 16 + l15;
          const float hv = h_s[row * H_LD + col];
          const float v  = sigmoidf_(acc[t][r]) + hv;
          h_s[row * H_LD + col] = v > 0.0f ? v : 0.0f;
        }
      }
    }
    __syncthreads();  // h updated; hw_t/w_t reads done -> next layer may restage
  }

  // ---------- readout: g = sum_n h; out = sigmoid(g @ W_fc + b_fc) ----------
  if (tid < H_SZ) {
    float s = 0.0f;
    for (int m = 0; m < N_SZ; ++m) s += h_s[m * H_LD + tid];
    red_s[tid] = s * W_fc[tid];
  }
  __syncthreads();
  if (tid == 0) {
    float t = 0.0f;
    #pragma unroll
    for (int i = 0; i < H_SZ; ++i) t += red_s[i];
    t += b_fc[0];
    out[b] = sigmoidf_(t);
  }
}

extern "C" void kernel_launch(void* const* d_in, const int* in_sizes, int n_in,
                              void* d_out, int out_size, void* d_ws, size_t ws_size,
                              hipStream_t stream) {
  (void)in_sizes; (void)n_in; (void)d_ws; (void)ws_size; (void)out_size;
  const float* node  = (const float*)d_in[0];
  const float* adj   = (const float*)d_in[1];
  const float* W_emb = (const float*)d_in[2];
  const float* W_gcn = (const float*)d_in[3];
  const float* b_gcn = (const float*)d_in[4];
  const float* W_fc  = (const float*)d_in[5];
  const float* b_fc  = (const float*)d_in[6];
  float* out = (float*)d_out;

  const size_t smem_bytes = (size_t)SMEM_FLOATS * sizeof(float); // ~150 KB

  hipFuncSetAttribute((const void*)gcn_fused_kernel,
                      hipFuncAttributeMaxDynamicSharedMemorySize,
                      (int)smem_bytes);

  gcn_fused_kernel<<<4096, 256, smem_bytes, stream>>>(
      node, adj, W_emb, W_gcn, b_gcn, W_fc, b_fc, out);
}